// NCC_vxm_730144441167
// MI455X (gfx1250) — compile-verified
//
#include <hip/hip_runtime.h>

typedef __attribute__((ext_vector_type(2))) float v2f;
typedef __attribute__((ext_vector_type(8))) float v8f;

#define NVOX 192
#define TILE 16
#define HALO 4
#define INTL 24                 // input tile edge (TILE + 2*HALO)
#define TI_OFF 0                // 24^3 = 13824 floats
#define TJ_OFF 13824
#define S1_OFF 27648            // 576 x 16 = 9216 floats
#define S2_OFF 36864            // 24 x 16 x 16 = 6144 floats
#define FIN_OFF 43008           // 5 x 16^3 = 20480 floats
#define SMEM_FLOATS 63488       // 253,952 bytes < 320 KB WGP LDS

// band matrix element: 1 if output o's 9-wide window covers input k (k in [o, o+8])
__device__ __forceinline__ float bandv(int k, int o) {
    return (k >= o && k <= o + 8) ? 1.0f : 0.0f;
}

__global__ __launch_bounds__(256) void ncc_kernel(const float* __restrict__ I,
                                                  const float* __restrict__ J,
                                                  float* __restrict__ out) {
    extern __shared__ float sm[];
    float* tI  = sm + TI_OFF;
    float* tJ  = sm + TJ_OFF;
    float* s1  = sm + S1_OFF;
    float* s2  = sm + S2_OFF;
    float* fin = sm + FIN_OFF;

    const int tid   = threadIdx.x;
    const int lane  = tid & 31;
    const int wv    = tid >> 5;          // wave id 0..7
    const int n     = lane & 15;         // matrix column (and A-row for stages 2/3)
    const int hi    = (lane >> 4) & 1;   // lane >= 16 ?
    const int khalf = hi * 2;            // K offset within a K=4 group
    const int z0 = blockIdx.z * TILE, y0 = blockIdx.y * TILE, x0 = blockIdx.x * TILE;

    // ---- cooperative load of 24^3 halo tiles (zero-padded at volume edges) ----
    for (int t = tid; t < INTL * INTL * INTL; t += 256) {
        int dz = t / (INTL * INTL);
        int r  = t - dz * INTL * INTL;
        int hy = r / INTL;
        int wx = r - hy * INTL;
        int gz = z0 + dz - HALO, gy = y0 + hy - HALO, gx = x0 + wx - HALO;
        float vi = 0.0f, vj = 0.0f;
        if ((unsigned)gz < NVOX && (unsigned)gy < NVOX && (unsigned)gx < NVOX) {
            int gi = (gz * NVOX + gy) * NVOX + gx;
            vi = I[gi];
            vj = J[gi];
        }
        tI[t] = vi;
        tJ[t] = vj;
    }
    __syncthreads();

    // channels: 0=I 1=J 2=I*I 3=J*J 4=I*J
    for (int c = 0; c < 5; ++c) {
        // ---- stage 1: box-sum along x.  D(16x16) = A_data(16x24) x S(24x16) ----
        // rows = flattened (d,h): 576 rows -> 36 chunks of 16
        for (int it = 0; it < 5; ++it) {
            int m = wv + 8 * it;                 // wave-uniform chunk id
            if (m < 36) {
                int row  = m * 16 + n;
                int dd   = row / INTL;
                int hh   = row - dd * INTL;
                int base = (dd * INTL + hh) * INTL;
                v8f acc = {0.f, 0.f, 0.f, 0.f, 0.f, 0.f, 0.f, 0.f};
#pragma unroll
                for (int kg = 0; kg < 6; ++kg) {
                    int k0 = kg * 4 + khalf;
                    float i0 = tI[base + k0], i1 = tI[base + k0 + 1];
                    float j0 = tJ[base + k0], j1 = tJ[base + k0 + 1];
                    v2f a, b;
                    switch (c) {
                        case 0:  a[0] = i0;      a[1] = i1;      break;
                        case 1:  a[0] = j0;      a[1] = j1;      break;
                        case 2:  a[0] = i0 * i0; a[1] = i1 * i1; break;
                        case 3:  a[0] = j0 * j0; a[1] = j1 * j1; break;
                        default: a[0] = i0 * j0; a[1] = i1 * j1; break;
                    }
                    b[0] = bandv(k0, n);
                    b[1] = bandv(k0 + 1, n);
                    acc = __builtin_amdgcn_wmma_f32_16x16x4_f32(
                        false, a, false, b, (short)0, acc, false, false);
                }
#pragma unroll
                for (int v = 0; v < 8; ++v) {
                    int mr = m * 16 + v + 8 * hi;      // absolute (d,h) row
                    s1[mr * 16 + n] = acc[v];
                }
            }
        }
        __syncthreads();

        // ---- stage 2: box-sum along h.  D = S^T(16x24) x s1_slice(24x16) ----
        for (int it = 0; it < 3; ++it) {
            int dd = wv + 8 * it;                // d-slice 0..23, wave-uniform
            v8f acc = {0.f, 0.f, 0.f, 0.f, 0.f, 0.f, 0.f, 0.f};
#pragma unroll
            for (int kg = 0; kg < 6; ++kg) {
                int k0 = kg * 4 + khalf;
                v2f a, b;
                a[0] = bandv(k0, n);             // A[m=n][k0]
                a[1] = bandv(k0 + 1, n);
                b[0] = s1[(dd * INTL + k0) * 16 + n];
                b[1] = s1[(dd * INTL + k0 + 1) * 16 + n];
                acc = __builtin_amdgcn_wmma_f32_16x16x4_f32(
                    false, a, false, b, (short)0, acc, false, false);
            }
#pragma unroll
            for (int v = 0; v < 8; ++v) {
                int ho = v + 8 * hi;             // output h row
                s2[(dd * 16 + ho) * 16 + n] = acc[v];
            }
        }
        __syncthreads();

        // ---- stage 3: box-sum along d.  D = S^T(16x24) x s2_col(24x16) ----
        for (int it = 0; it < 2; ++it) {
            int hh = wv + 8 * it;                // h-slice 0..15, wave-uniform
            v8f acc = {0.f, 0.f, 0.f, 0.f, 0.f, 0.f, 0.f, 0.f};
#pragma unroll
            for (int kg = 0; kg < 6; ++kg) {
                int k0 = kg * 4 + khalf;
                v2f a, b;
                a[0] = bandv(k0, n);
                a[1] = bandv(k0 + 1, n);
                b[0] = s2[(k0 * 16 + hh) * 16 + n];
                b[1] = s2[((k0 + 1) * 16 + hh) * 16 + n];
                acc = __builtin_amdgcn_wmma_f32_16x16x4_f32(
                    false, a, false, b, (short)0, acc, false, false);
            }
#pragma unroll
            for (int v = 0; v < 8; ++v) {
                int dout = v + 8 * hi;           // output d row
                fin[c * 4096 + (dout * 16 + hh) * 16 + n] = acc[v];
            }
        }
        __syncthreads();
    }

    // ---- pointwise NCC combine + local accumulation (16 voxels / thread) ----
    const float W3  = 729.0f;                    // 9^3
    const float iW3 = 1.0f / 729.0f;
    float local = 0.0f;
#pragma unroll
    for (int q = 0; q < 16; ++q) {
        int idx  = tid + 256 * q;
        float Is = fin[idx];
        float Js = fin[4096 + idx];
        float I2 = fin[2 * 4096 + idx];
        float J2 = fin[3 * 4096 + idx];
        float IJ = fin[4 * 4096 + idx];
        float uI = Is * iW3, uJ = Js * iW3;
        float cross = IJ - uJ * Is - uI * Js + uI * uJ * W3;
        float Iv = I2 - 2.0f * uI * Is + uI * uI * W3;
        float Jv = J2 - 2.0f * uJ * Js + uJ * uJ * W3;
        local += cross * cross / (Iv * Jv + 1e-5f);
    }

    // ---- block reduction (reuse s1 scratch) + single atomic per block ----
    float* red = s1;
    red[tid] = local;
    __syncthreads();
#pragma unroll
    for (int s = 128; s > 0; s >>= 1) {
        if (tid < s) red[tid] += red[tid + s];
        __syncthreads();
    }
    if (tid == 0) {
        const float scale = -1.0f / 7077888.0f;  // -mean over 192^3
        atomicAdd(out, red[0] * scale);
    }
}

__global__ void ncc_zero_kernel(float* out) {
    out[0] = 0.0f;
}

extern "C" void kernel_launch(void* const* d_in, const int* in_sizes, int n_in,
                              void* d_out, int out_size, void* d_ws, size_t ws_size,
                              hipStream_t stream) {
    const float* I = (const float*)d_in[0];   // y_true
    const float* J = (const float*)d_in[1];   // y_pred
    float* out = (float*)d_out;

    ncc_zero_kernel<<<1, 1, 0, stream>>>(out);

    dim3 grid(NVOX / TILE, NVOX / TILE, NVOX / TILE);   // 12^3 tiles
    ncc_kernel<<<grid, 256, SMEM_FLOATS * sizeof(float), stream>>>(I, J, out);
}